// ReLU6_47940424958602
// MI455X (gfx1250) — compile-verified
//
#include <hip/hip_runtime.h>

#define N_NEURONS 8192
#define M_DIM (N_NEURONS + 1)
#define EPS_ALPHA 1e-5f

#define GLOBAL_AS __attribute__((address_space(1)))
#define LDS_AS    __attribute__((address_space(3)))

typedef int   v4i __attribute__((vector_size(16)));   // matches builtin param type
typedef __attribute__((ext_vector_type(4))) float float4v;

// ---------------------------------------------------------------------------
// Zero-fill of the two (M x M) matrices: ~0.54 GB of pure stores.
// CDNA5 async LDS->global B128 stores (ASYNCcnt path): each wave32
// instruction streams a contiguous 512B of zeros from LDS with no data-VGPR
// traffic. Each wave owns 2KB chunks; 4 stores per loop trip via the 24-bit
// immediate offset (0/512/1024/1536) to keep the loop HBM-limited, not
// issue-limited. Fallback: nontemporal 128-bit VGPR stores.
// ---------------------------------------------------------------------------
__global__ void fill_zero_async(float* __restrict__ base, long chunk_count) {
    __shared__ float lz[128];  // 512 B of zeros, one 16B chunk per lane
    if (threadIdx.x < 128) lz[threadIdx.x] = 0.0f;
    __syncthreads();

    const int  wpb     = blockDim.x >> 5;                       // waves per block
    const long nwaves  = (long)gridDim.x * wpb;
    const long wave_id = (long)blockIdx.x * wpb + (threadIdx.x >> 5);
    const int  lane    = threadIdx.x & 31;

#if __has_builtin(__builtin_amdgcn_global_store_async_from_lds_b128)
    LDS_AS v4i* lds_src = (LDS_AS v4i*)(&lz[lane * 4]);

    unsigned long long p    = (unsigned long long)base
                            + (unsigned long long)wave_id * 2048ull
                            + (unsigned long long)(lane * 16);
    unsigned long long end  = (unsigned long long)base
                            + (unsigned long long)chunk_count * 2048ull;
    unsigned long long step = (unsigned long long)nwaves * 2048ull;

    for (; p < end; p += step) {
        GLOBAL_AS v4i* g = (GLOBAL_AS v4i*)p;
        // (global dst, lds src, imm byte offset, imm cpol)
        __builtin_amdgcn_global_store_async_from_lds_b128(g, lds_src, 0,    0);
        __builtin_amdgcn_global_store_async_from_lds_b128(g, lds_src, 512,  0);
        __builtin_amdgcn_global_store_async_from_lds_b128(g, lds_src, 1024, 0);
        __builtin_amdgcn_global_store_async_from_lds_b128(g, lds_src, 1536, 0);
    }
#if __has_builtin(__builtin_amdgcn_s_wait_asynccnt)
    __builtin_amdgcn_s_wait_asynccnt(0);
#endif
#else
    float4v z = {0.0f, 0.0f, 0.0f, 0.0f};
    long vec4_count = chunk_count * 128;   // 128 float4s per 2KB chunk
    long idx    = (long)blockIdx.x * blockDim.x + threadIdx.x;
    long stride = (long)gridDim.x * blockDim.x;
    for (; idx < vec4_count; idx += stride) {
        __builtin_nontemporal_store(z, (float4v*)(base + 4 * idx));
    }
#endif
}

// ---------------------------------------------------------------------------
// Elementwise ReLU6 DeepPoly relaxation + sparse scatter into the matrices.
// Masks mA..mE are mutually exclusive, so nested selects reproduce the
// reference's mask-multiply sums exactly (denominators are guarded, so all
// unselected terms are finite and contribute 0).
// ---------------------------------------------------------------------------
__global__ void relax_scatter(const float* __restrict__ lower,
                              const float* __restrict__ upper,
                              float* __restrict__ out) {
    int i = blockIdx.x * blockDim.x + threadIdx.x;
    if (i >= N_NEURONS) return;

    float l = lower[i];
    float u = upper[i];

    float den_ul = (u > l)    ? (u - l)    : 1.0f;
    float den_6l = (l < 6.0f) ? (6.0f - l) : 1.0f;
    float u_safe = (u > 0.0f) ? u          : 1.0f;

    bool mA = (u > 0.0f) && (u <= 6.0f) && (l >= 0.0f);
    bool mB = (u > 0.0f) && (u <= 6.0f) && (l < 0.0f);
    bool mC = (u > 6.0f) && (l <= 0.0f);
    bool mD = (u > 6.0f) && (l > 0.0f) && (l <= 6.0f);
    bool mE = (l > 6.0f);

    float alpha_B = (u < -l) ? EPS_ALPHA : 1.0f;
    float lam_B   = u / den_ul;
    float aU_C    = ((u - 6.0f) < (6.0f - l)) ? (6.0f / den_6l) : EPS_ALPHA;
    float aL_C    = (u < -l) ? EPS_ALPHA : (6.0f / u_safe);
    float aU_D    = ((u - 6.0f) < (6.0f - l)) ? 1.0f : EPS_ALPHA;
    float aL_D    = (6.0f - l) / den_ul;

    float diag_u = mA ? 1.0f    : mB ? lam_B   : mC ? aU_C : mD ? aU_D : 0.0f;
    float diag_l = mA ? 1.0f    : mB ? alpha_B : mC ? aL_C : mD ? aL_D : 0.0f;
    float bias_u = mB ? (-lam_B * l)
                 : mC ? (6.0f * (1.0f - aU_C))
                 : mD ? (6.0f * (1.0f - aU_D))
                 : mE ? 6.0f : 0.0f;
    float bias_l = mD ? (l * (1.0f - aL_D)) : mE ? 6.0f : 0.0f;
    float cu     = mA ? u : mB ? u
                 : mC ? (6.0f + aU_C * (u - 6.0f))
                 : mD ? (6.0f + aU_D * (u - 6.0f))
                 : mE ? 6.0f : 0.0f;
    float cl     = mA ? l : mB ? (alpha_B * l) : mC ? (aL_C * l)
                 : mD ? l : mE ? 6.0f : 0.0f;

    // Output layout: [cl (N)] [cu (N)] [A_l (M*M)] [A_u (M*M)]
    float* cl_out = out;
    float* cu_out = out + N_NEURONS;
    float* Al     = out + 2 * (size_t)N_NEURONS;
    float* Au     = Al + (size_t)M_DIM * (size_t)M_DIM;

    cl_out[i] = cl;
    cu_out[i] = cu;

    size_t diag_idx = (size_t)i * M_DIM + i;
    size_t last_row = (size_t)(M_DIM - 1) * M_DIM + i;
    Al[diag_idx] = diag_l;
    Au[diag_idx] = diag_u;
    Al[last_row] = bias_l;
    Au[last_row] = bias_u;

    if (i == 0) {
        Al[(size_t)M_DIM * M_DIM - 1] = 1.0f;
        Au[(size_t)M_DIM * M_DIM - 1] = 1.0f;
    }
}

extern "C" void kernel_launch(void* const* d_in, const int* in_sizes, int n_in,
                              void* d_out, int out_size, void* d_ws, size_t ws_size,
                              hipStream_t stream) {
    const float* lower = (const float*)d_in[0];
    const float* upper = (const float*)d_in[1];
    float* out = (float*)d_out;

    const long M = M_DIM;
    const long mat_elems  = 2L * M * M;              // 134,250,498 floats
    const long vec4_cnt   = mat_elems / 4;           // 33,562,624 float4s
    const long chunk_cnt  = vec4_cnt / 128;          // 262,208 x 2KB chunks (exact)
    // 2-element scalar tail lies in A_u's last row -> rewritten by scatter.
    float* mats = out + 2 * (size_t)N_NEURONS;

    // 1) Bulk zero-fill (store-bandwidth-bound, async LDS->global path).
    fill_zero_async<<<8192, 256, 0, stream>>>(mats, chunk_cnt);

    // 2) Elementwise relaxation + sparse scatter (ordered after the fill).
    relax_scatter<<<(N_NEURONS + 255) / 256, 256, 0, stream>>>(lower, upper, out);
}